// SemanticDecoupling_83820581749246
// MI455X (gfx1250) — compile-verified
//
#include <hip/hip_runtime.h>

// Sizes (fixed by the reference)
#define B_   4
#define S_   14
#define C_   2048
#define K_   80
#define WD_  300
#define SEM_ 1024
#define HW_  196              // S*S
#define BP_  784              // B*HW
#define CS_  28672            // C*S

typedef __attribute__((ext_vector_type(2))) float v2f;
typedef __attribute__((ext_vector_type(4))) float v4f;
typedef __attribute__((ext_vector_type(8))) float v8f;

__device__ __forceinline__ float fast_tanh(float v) {
    // tanh(x) = 1 - 2/(1 + e^{2x});  e^{2x} = exp2(x * 2*log2(e))
    float e = __builtin_amdgcn_exp2f(v * 2.8853900817779268f);
    return 1.0f - 2.0f * __builtin_amdgcn_rcpf(e + 1.0f);
}

// K0a: word_p[k][s] = sum_w word_features[k,w] * W_word[s,w]   (80 x 1024)
__global__ void sd_prep_wordp(const float* __restrict__ wf,
                              const float* __restrict__ Ww,
                              float* __restrict__ wordp) {
    int s = blockIdx.x * 256 + threadIdx.x;   // grid.x = 4
    int k = blockIdx.y;                       // 80
    float acc = 0.f;
    for (int w = 0; w < WD_; ++w)
        acc = fmaf(wf[k * WD_ + w], Ww[s * WD_ + w], acc);
    wordp[k * SEM_ + s] = acc;
}

// K0b: w_eff[s] = sum_t W2[t]*W1[t,s];  b_eff = W2·b1 + b2
__global__ void sd_prep_weff(const float* __restrict__ W1,
                             const float* __restrict__ b1,
                             const float* __restrict__ W2,
                             const float* __restrict__ b2,
                             float* __restrict__ weff,
                             float* __restrict__ beff) {
    int s = blockIdx.x * 256 + threadIdx.x;   // 4 blocks -> 1024 threads
    float acc = 0.f;
    for (int t = 0; t < SEM_; ++t)
        acc = fmaf(W2[t], W1[t * SEM_ + s], acc);
    weff[s] = acc;
    if (s == 0) {
        float bb = b2[0];
        for (int t = 0; t < SEM_; ++t) bb = fmaf(W2[t], b1[t], bb);
        beff[0] = bb;
    }
}

// K1: x[b,p,c] = image_map[b, i2, i3, i1] with f=p*C+c; i1=f/(C*S); i2=(f%(C*S))/S; i3=f%S
__global__ void sd_gather_x(const float* __restrict__ im, float* __restrict__ x) {
    unsigned gid = blockIdx.x * 256u + threadIdx.x;   // exactly B*HW*C = 1,605,632
    unsigned b  = gid / (HW_ * C_);
    unsigned f  = gid % (HW_ * C_);
    unsigned i1 = f / CS_;
    unsigned r  = f % CS_;
    unsigned i2 = r / S_;
    unsigned i3 = r % S_;
    x[gid] = im[b * (C_ * S_ * S_) + i2 * (S_ * S_) + i3 * S_ + i1];
}

// K2: img_p (784 x 1024) = x (784 x 2048) @ W_img^T (2048 x 1024), fp32 WMMA 16x16x4
__global__ void sd_imgp_gemm(const float* __restrict__ x,
                             const float* __restrict__ Wimg,
                             float* __restrict__ imgp) {
    unsigned wid  = (blockIdx.x * blockDim.x + threadIdx.x) >> 5;  // 0..3135
    unsigned lane = threadIdx.x & 31u;
    unsigned half = lane >> 4, l16 = lane & 15u;
    unsigned tm = wid >> 6;        // 49 M-tiles
    unsigned tn = wid & 63u;       // 64 N-tiles
    unsigned m0 = tm * 16u, n0 = tn * 16u;
    const float* xrow = x    + (m0 + l16) * C_ + 2u * half;
    const float* wrow = Wimg + (n0 + l16) * C_ + 2u * half;
    v8f acc = {};
    #pragma unroll 8
    for (int kk = 0; kk < C_; kk += 4) {
        v2f a = *(const v2f*)(xrow + kk);
        v2f b = *(const v2f*)(wrow + kk);
        acc = __builtin_amdgcn_wmma_f32_16x16x4_f32(false, a, false, b,
                                                    (short)0, acc, false, false);
    }
    float* orow = imgp + n0 + l16;
    #pragma unroll
    for (int v = 0; v < 8; ++v)
        orow[(m0 + v + 8u * half) * SEM_] = acc[v];
}

// K3: attn[b,p,k] = softmax_k( b_eff + sum_s tanh(img_p[bp,s]*word_p[k,s]) * w_eff[s] )
__global__ void sd_attn(const float* __restrict__ imgp,
                        const float* __restrict__ wordp,
                        const float* __restrict__ weff,
                        const float* __restrict__ beff,
                        float* __restrict__ attn_out) {
    __shared__ float s_img[SEM_];
    __shared__ float s_weff[SEM_];
    __shared__ float s_logit[K_];
    int bp  = blockIdx.x;        // 784
    int tid = threadIdx.x;       // 256 = 8 waves
    for (int i = tid; i < SEM_; i += 256) {
        s_img[i]  = imgp[bp * SEM_ + i];
        s_weff[i] = weff[i];
    }
    __syncthreads();
    int wave = tid >> 5, lane = tid & 31;
    float be = beff[0];
    for (int j = 0; j < K_ / 8; ++j) {         // 10 classes per wave
        int k = wave * (K_ / 8) + j;
        const float* wp = wordp + k * SEM_;
        float acc = 0.f;
        for (int i = lane; i < SEM_; i += 32)
            acc = fmaf(fast_tanh(s_img[i] * wp[i]), s_weff[i], acc);
        for (int off = 16; off > 0; off >>= 1)
            acc += __shfl_xor(acc, off, 32);
        if (lane == 0) s_logit[k] = acc + be;
    }
    __syncthreads();
    if (tid < K_) {
        float m = -1e30f;
        for (int j = 0; j < K_; ++j) m = fmaxf(m, s_logit[j]);
        float sum = 0.f;
        for (int j = 0; j < K_; ++j)
            sum += __builtin_amdgcn_exp2f((s_logit[j] - m) * 1.4426950408889634f);
        float e = __builtin_amdgcn_exp2f((s_logit[tid] - m) * 1.4426950408889634f);
        attn_out[bp * K_ + tid] = e * __builtin_amdgcn_rcpf(sum);
    }
}

// K4: fmwc[b,p,k,c] = x[b,p,c] * attn[b,p,k]   (514 MB, NT stores)
__global__ void sd_fmwc(const float* __restrict__ x,
                        const float* __restrict__ attn,
                        float* __restrict__ fmwc) {
    unsigned gid = blockIdx.x * 256u + threadIdx.x;   // BP_*K_*512 = 32,112,640
    unsigned c4  = gid & 511u;                        // 2048/4
    unsigned t   = gid >> 9;                          // bp*80 + k
    unsigned k   = t % K_;
    unsigned bp  = t / K_;
    float a  = attn[t];
    v4f  xv = *(const v4f*)(x + bp * C_ + c4 * 4u);
    v4f  o  = xv * a;
    __builtin_nontemporal_store(o, (v4f*)(fmwc + (unsigned long long)t * C_ + c4 * 4u));
}

// K5: semantic[b,k,c] = sum_p attn[b,p,k] * x[b,p,c]  -> per-b GEMM 80x2048x196, fp32 WMMA
__global__ void sd_sem_gemm(const float* __restrict__ x,
                            const float* __restrict__ attn,
                            float* __restrict__ sem) {
    unsigned wid  = (blockIdx.x * blockDim.x + threadIdx.x) >> 5;  // 0..2559
    unsigned lane = threadIdx.x & 31u;
    unsigned half = lane >> 4, l16 = lane & 15u;
    unsigned b  = wid / 640u;          // 4
    unsigned r  = wid % 640u;
    unsigned tm = r >> 7;              // 5 M-tiles (80 rows)
    unsigned tn = r & 127u;            // 128 N-tiles (2048 cols)
    unsigned m0 = tm * 16u, n0 = tn * 16u;
    const float* ab = attn + b * (HW_ * K_);
    const float* xb = x    + b * (HW_ * C_);
    v8f acc = {};
    for (int kk = 0; kk < HW_; kk += 4) {          // 196 = 49 * 4
        int kd = kk + 2 * (int)half;
        v2f a, bv;
        a.x  = ab[kd * K_ + m0 + l16];
        a.y  = ab[(kd + 1) * K_ + m0 + l16];
        bv.x = xb[kd * C_ + n0 + l16];
        bv.y = xb[(kd + 1) * C_ + n0 + l16];
        acc = __builtin_amdgcn_wmma_f32_16x16x4_f32(false, a, false, bv,
                                                    (short)0, acc, false, false);
    }
    float* ob = sem + b * (K_ * C_) + n0 + l16;
    #pragma unroll
    for (int v = 0; v < 8; ++v)
        ob[(m0 + v + 8u * half) * C_] = acc[v];
}

extern "C" void kernel_launch(void* const* d_in, const int* in_sizes, int n_in,
                              void* d_out, int out_size, void* d_ws, size_t ws_size,
                              hipStream_t stream) {
    const float* image_map     = (const float*)d_in[0];
    const float* word_features = (const float*)d_in[1];
    const float* W_img         = (const float*)d_in[2];
    const float* W_word        = (const float*)d_in[3];
    const float* W1            = (const float*)d_in[4];
    const float* b1            = (const float*)d_in[5];
    const float* W2            = (const float*)d_in[6];
    const float* b2            = (const float*)d_in[7];

    float* ws    = (float*)d_ws;
    float* x     = ws;                    // 1,605,632 floats
    float* imgp  = x + 1605632;           //   802,816
    float* wordp = imgp + 802816;         //    81,920
    float* weff  = wordp + 81920;         //     1,024
    float* beff  = weff + 1024;           //         1

    float* out  = (float*)d_out;
    float* sem  = out;                            //     655,360
    float* fmwc = out + 655360;                   // 128,450,560
    float* attn = out + 655360 + 128450560;       //      62,720

    sd_prep_wordp<<<dim3(4, 80), 256, 0, stream>>>(word_features, W_word, wordp);
    sd_prep_weff <<<4,            256, 0, stream>>>(W1, b1, W2, b2, weff, beff);
    sd_gather_x  <<<6272,         256, 0, stream>>>(image_map, x);
    sd_imgp_gemm <<<392,          256, 0, stream>>>(x, W_img, imgp);     // 3136 waves
    sd_attn      <<<784,          256, 0, stream>>>(imgp, wordp, weff, beff, attn);
    sd_fmwc      <<<125440,       256, 0, stream>>>(x, attn, fmwc);
    sd_sem_gemm  <<<320,          256, 0, stream>>>(x, attn, sem);       // 2560 waves
}